// GeometricWorldModel_10617159155907
// MI455X (gfx1250) — compile-verified
//
#include <hip/hip_runtime.h>
#include <math.h>

// ---- problem constants (from reference setup_inputs) ----
#define B_ROWS   2048
#define D_DIM    256
#define K_CH     64
#define A_DIM    64
#define DM       1024
#define LDS_PITCH 260   // 256 + 4 pad: keeps float4 alignment, spreads LDS banks
#define TILES_B  4      // 16-row b-tiles per wave (M register blocking)
#define WAVES_K  4      // waves per block, each owning one chart k

typedef __attribute__((ext_vector_type(2))) float v2f;
typedef __attribute__((ext_vector_type(8))) float v8f;

// ---------------------------------------------------------------------------
// Kernel 1: ball-project chart centers -> c_ws [K,D], c2_ws [K]
// ---------------------------------------------------------------------------
__global__ __launch_bounds__(256)
void gwm_centers(const float* __restrict__ cc,
                 float* __restrict__ c_ws, float* __restrict__ c2_ws)
{
    __shared__ float red[256];
    const int k = blockIdx.x;
    const int d = threadIdx.x;
    const float x = cc[k * D_DIM + d];
    red[d] = x * x;
    __syncthreads();
    for (int s = 128; s > 0; s >>= 1) {
        if (d < s) red[d] += red[d + s];
        __syncthreads();
    }
    const float sumsq = red[0];
    const float n     = sqrtf(sumsq);
    const float scale = fminf(1.0f, (1.0f - 1e-5f) / fmaxf(n, 1e-12f));
    c_ws[k * D_DIM + d] = x * scale;
    if (d == 0) c2_ws[k] = sumsq * scale * scale;
}

// ---------------------------------------------------------------------------
// Kernel 2: sigma (power iteration), T2[a,k] = (act_weight @ proj_W^T)/(A*sigma),
//           bias[k] = proj_b[k] + (pos_mean . proj_W[k,:]) / sigma
// ---------------------------------------------------------------------------
__global__ __launch_bounds__(1024)
void gwm_actprep(const float* __restrict__ act_weight,
                 const float* __restrict__ act_pos,
                 const float* __restrict__ proj_W,
                 const float* __restrict__ proj_b,
                 float* __restrict__ T2, float* __restrict__ bias)
{
    __shared__ float pm[DM];
    __shared__ float v[DM];
    __shared__ float u[K_CH];
    __shared__ float red[DM];
    __shared__ float s_sigma, s_unorm;
    const int tid = threadIdx.x;

    {
        float s = 0.f;
        for (int a = 0; a < A_DIM; ++a) s += act_pos[a * DM + tid];
        pm[tid] = s * (1.0f / (float)A_DIM);
    }
    v[tid] = 1.0f;
    __syncthreads();

    for (int it = 0; it < 16; ++it) {
        red[tid] = v[tid] * v[tid];
        __syncthreads();
        for (int st = 512; st > 0; st >>= 1) {
            if (tid < st) red[tid] += red[tid + st];
            __syncthreads();
        }
        const float vn = sqrtf(red[0]) + 1e-20f;
        __syncthreads();
        v[tid] /= vn;
        __syncthreads();
        if (tid < K_CH) {
            float acc = 0.f;
            const float* wr = proj_W + tid * DM;
            for (int m = 0; m < DM; ++m) acc += wr[m] * v[m];
            u[tid] = acc;
        }
        __syncthreads();
        if (tid == 0) {
            float un = 0.f;
            for (int kk = 0; kk < K_CH; ++kk) un += u[kk] * u[kk];
            un = sqrtf(un);
            s_sigma = un;            // ||W v|| with v unit -> sigma estimate
            s_unorm = un + 1e-20f;
        }
        __syncthreads();
        {
            float acc = 0.f;
            for (int kk = 0; kk < K_CH; ++kk) acc += proj_W[kk * DM + tid] * u[kk];
            v[tid] = acc / s_unorm;
        }
        __syncthreads();
    }

    const float inv_sigma = 1.0f / (s_sigma + 1e-12f);

    if (tid < K_CH) {
        float acc = 0.f;
        const float* wr = proj_W + tid * DM;
        for (int m = 0; m < DM; ++m) acc += pm[m] * wr[m];
        bias[tid] = proj_b[tid] + acc * inv_sigma;
    }
    const float scaleT = inv_sigma * (1.0f / (float)A_DIM);
    for (int idx = tid; idx < A_DIM * K_CH; idx += 1024) {
        const int a  = idx >> 6;
        const int kk = idx & 63;
        float acc = 0.f;
        const float* awr = act_weight + a * DM;
        const float* pwr = proj_W + kk * DM;
        for (int m = 0; m < DM; ++m) acc += awr[m] * pwr[m];
        T2[a * K_CH + kk] = acc * scaleT;
    }
}

// ---------------------------------------------------------------------------
// Kernel 3 (dominant): block = 4 waves sharing one 64x256 Z tile in LDS.
//   Wave w owns chart k = blockIdx.y*4 + w and 4 register-blocked 16-row
//   b-tiles: each Q_k B-fragment is reused by 4 WMMAs (4x less L2 traffic).
//   gamma = z^T Q_k z fused on the accumulator fragments; hyperbolic base
//   and collapsed action terms finished in the epilogue.
// ---------------------------------------------------------------------------
__global__ __launch_bounds__(32 * WAVES_K)
void gwm_main(const float* __restrict__ z, const float* __restrict__ action,
              const float* __restrict__ qg, const float* __restrict__ c_ws,
              const float* __restrict__ c2_ws, const float* __restrict__ T2,
              const float* __restrict__ bias, float* __restrict__ out)
{
    __shared__ __align__(16) float Zl[64 * LDS_PITCH];   // ~65 KB, shared by 4 waves
    __shared__ float gbuf[WAVES_K][64];
    const int tid  = threadIdx.x;
    const int lane = tid & 31;
    const int wave = tid >> 5;
    const int b0   = blockIdx.x * 64;
    const int k    = blockIdx.y * WAVES_K + wave;

    // cooperative stage of the 64x256 Z tile (float4, padded pitch)
    const float* zsrc = z + (size_t)b0 * D_DIM;
    for (int t = tid; t < (64 * D_DIM) / 4; t += 32 * WAVES_K) {
        const int f   = t * 4;
        const int row = f >> 8;     // /256
        const int col = f & 255;
        *(float4*)(&Zl[row * LDS_PITCH + col]) = *(const float4*)(zsrc + f);
    }
    __syncthreads();

    const int lo = lane & 15;       // A: M row / B,D: N col
    const int hi = lane >> 4;       // selects K pair (A/B) and M half (D)
    const float* qk = qg + (size_t)k * (D_DIM * D_DIM);

    float psum[TILES_B][8] = {};

    for (int jt = 0; jt < 16; ++jt) {
        const int j0 = jt * 16;
        v8f acc[TILES_B];
        #pragma unroll
        for (int t = 0; t < TILES_B; ++t)
            acc[t] = (v8f){0.f, 0.f, 0.f, 0.f, 0.f, 0.f, 0.f, 0.f};

        #pragma unroll 2
        for (int i0 = 0; i0 < D_DIM; i0 += 4) {
            const int ia = i0 + 2 * hi;
            v2f b;                                   // B[K=ia..ia+1][N=lo]
            b.x = qk[ia * D_DIM + j0 + lo];
            b.y = qk[(ia + 1) * D_DIM + j0 + lo];
            #pragma unroll
            for (int t = 0; t < TILES_B; ++t) {      // reuse B across 4 M-tiles
                v2f a;                               // A[M=t*16+lo][K=ia..ia+1]
                a.x = Zl[(t * 16 + lo) * LDS_PITCH + ia];
                a.y = Zl[(t * 16 + lo) * LDS_PITCH + ia + 1];
                acc[t] = __builtin_amdgcn_wmma_f32_16x16x4_f32(
                    false, a, false, b, (short)0, acc[t], false, false);
            }
        }
        // fuse second contraction: gamma += D[m][n] * z[b0+m][j0+n]
        // D layout: lane (n + 16*hi), vgpr r -> element (m = r + 8*hi, n = lo)
        #pragma unroll
        for (int t = 0; t < TILES_B; ++t)
            #pragma unroll
            for (int r = 0; r < 8; ++r)
                psum[t][r] += acc[t][r] *
                    Zl[(t * 16 + r + 8 * hi) * LDS_PITCH + j0 + lo];
    }

    // reduce over n (16 lanes within each half; masks 1/2/4/8 stay in-half)
    #pragma unroll
    for (int t = 0; t < TILES_B; ++t) {
        #pragma unroll
        for (int r = 0; r < 8; ++r) {
            float vv = psum[t][r];
            vv += __shfl_xor(vv, 1, 32);
            vv += __shfl_xor(vv, 2, 32);
            vv += __shfl_xor(vv, 4, 32);
            vv += __shfl_xor(vv, 8, 32);
            psum[t][r] = vv;
        }
        if (lo == 0) {
            #pragma unroll
            for (int r = 0; r < 8; ++r)
                gbuf[wave][t * 16 + 8 * hi + r] = psum[t][r];
        }
    }
    __syncthreads();

    // epilogue: lanes 0..15 of each wave finish 4 rows (m = t*16 + lane)
    if (lane < 16) {
        const float c2k = c2_ws[k];
        const float* crow = c_ws + k * D_DIM;
        for (int t = 0; t < TILES_B; ++t) {
            const int m = t * 16 + lane;
            float z2 = 0.f, zc = 0.f;
            for (int j = 0; j < D_DIM; ++j) {
                const float zv = Zl[m * LDS_PITCH + j];
                z2 += zv * zv;
                zc += zv * crow[j];
            }
            const float sqd   = fmaxf(z2 - 2.f * zc + c2k, 0.f);
            const float denom = (1.f - z2) * (1.f - c2k);
            float arg = 1.f + 2.f * sqd / fmaxf(denom, 1e-12f);
            arg = fmaxf(arg, 1.f + 1e-7f);
            const float dist = logf(arg + sqrtf(arg * arg - 1.f));  // arccosh
            const float lam  = 2.f / fmaxf(1.f - z2, 1e-6f);
            const float tau  = lam * 0.0625f;                       // /sqrt(256)
            const float base = -dist / (tau + 1e-8f);

            float actv = bias[k];
            const float* arow = action + (size_t)(b0 + m) * A_DIM;
            #pragma unroll 8
            for (int a = 0; a < A_DIM; ++a)
                actv += arow[a] * T2[a * K_CH + k];

            out[(size_t)(b0 + m) * K_CH + k] = base + gbuf[wave][m] + actv;
        }
    }
}

// ---------------------------------------------------------------------------
extern "C" void kernel_launch(void* const* d_in, const int* in_sizes, int n_in,
                              void* d_out, int out_size, void* d_ws, size_t ws_size,
                              hipStream_t stream) {
    const float* z      = (const float*)d_in[0];  // [2048,256]
    const float* action = (const float*)d_in[1];  // [2048,64]
    // d_in[2] = rw : unused by the reference
    const float* cc     = (const float*)d_in[3];  // [64,256]
    const float* qg     = (const float*)d_in[4];  // [64,256,256]
    const float* aw     = (const float*)d_in[5];  // [64,1024]
    const float* ap     = (const float*)d_in[6];  // [64,1024]
    const float* pW     = (const float*)d_in[7];  // [64,1024]
    const float* pb     = (const float*)d_in[8];  // [64]
    float* out = (float*)d_out;                   // [2048,64]

    float* ws    = (float*)d_ws;
    float* c_ws  = ws;               // 16384 floats
    float* c2_ws = ws + 16384;       // 64
    float* T2    = ws + 16448;       // 4096
    float* bias  = ws + 20544;       // 64    (total ~82 KB)

    gwm_centers<<<dim3(K_CH), dim3(256), 0, stream>>>(cc, c_ws, c2_ws);
    gwm_actprep<<<dim3(1), dim3(1024), 0, stream>>>(aw, ap, pW, pb, T2, bias);
    gwm_main<<<dim3(B_ROWS / 64, K_CH / WAVES_K), dim3(32 * WAVES_K), 0, stream>>>(
        z, action, qg, c_ws, c2_ws, T2, bias, out);
}